// NequIPNBodyNetSimple_13950053777580
// MI455X (gfx1250) — compile-verified
//
#include <hip/hip_runtime.h>
#include <hip/hip_bf16.h>

// NequIP-style N-body net, B=8, N=256, C=2, H=128, L=4.
// Algebraic refactor: P=h@w1_a+b1, Q=h@w1_b, pre_ij=P_i+Q_j+d2_ij*w1_d,
// agg_i = (sum_{j!=i} silu(pre_ij)) @ w2 + 255*b2  (mask & @w2 commute).
// GEMMs use V_WMMA_F32_16X16X4_F32 (exact f32), O(N^2) part is fused VALU.

typedef __attribute__((ext_vector_type(2))) float v2f;
typedef __attribute__((ext_vector_type(8))) float v8f;

#define Hd   128
#define Nn   256
#define Bb   8
#define ROWS (Bb * Nn)   // 2048

// ---------------------------------------------------------------------------
// f32 WMMA: D(16x16) = A(16x4) x B(4x16) + C. 8-arg f32 pattern.
static __device__ __forceinline__ v8f wmma_f32(v2f a, v2f b, v8f c) {
  return __builtin_amdgcn_wmma_f32_16x16x4_f32(
      /*neg_a=*/false, a, /*neg_b=*/false, b,
      /*c_mod=*/(short)0, c, /*reuse_a=*/false, /*reuse_b=*/false);
}

// A fragment (16x4, row-major source, leading dim lda):
// lanes 0-15: M=lane, K=k0+{0,1}; lanes 16-31: M=lane-16, K=k0+{2,3}.
static __device__ __forceinline__ v2f load_a(const float* __restrict__ A,
                                             int lda, int rowBase, int k0,
                                             int lane16, int laneHalf) {
  const float* p = A + (size_t)(rowBase + lane16) * lda + k0 + 2 * laneHalf;
  v2f a; a.x = p[0]; a.y = p[1]; return a;
}

// B fragment (4x16, row-major source, leading dim ldb):
// lanes 0-15: N=colBase+lane, K=k0+{0,1}; lanes 16-31: K=k0+{2,3}.
static __device__ __forceinline__ v2f load_b(const float* __restrict__ Bm,
                                             int ldb, int k0, int colBase,
                                             int lane16, int laneHalf) {
  const float* p = Bm + (size_t)(k0 + 2 * laneHalf) * ldb + colBase + lane16;
  v2f b; b.x = p[0]; b.y = p[ldb]; return b;
}

// ---------------------------------------------------------------------------
// h = state @ embed_w + embed_b    (K=4, trivial)
__global__ __launch_bounds__(Hd) void embed_kernel(
    const float* __restrict__ state, const float* __restrict__ ew,
    const float* __restrict__ eb, float* __restrict__ h) {
  int row = blockIdx.x;          // 0..2047
  int t = threadIdx.x;           // channel 0..127
  const float* s = state + (size_t)row * 4;
  float acc = eb[t];
#pragma unroll
  for (int k = 0; k < 4; ++k) acc = fmaf(s[k], ew[k * Hd + t], acc);
  h[(size_t)row * Hd + t] = acc;
}

// ---------------------------------------------------------------------------
// P = h @ w1[0:128] + b1 ; Q = h @ w1[128:256].  One wave per 16x16 tile.
__global__ __launch_bounds__(128) void pq_gemm_kernel(
    const float* __restrict__ h, const float* __restrict__ w1l,
    const float* __restrict__ b1l, float* __restrict__ P,
    float* __restrict__ Q) {
  int wave = (blockIdx.x * blockDim.x + threadIdx.x) >> 5;
  int lane = threadIdx.x & 31;
  int lane16 = lane & 15, laneHalf = lane >> 4;
  int rowBase = (wave >> 3) * 16;   // 128 M-tiles
  int colBase = (wave & 7) * 16;    // 8 N-tiles
  v8f cP, cQ;
  float bp = b1l[colBase + lane16];
#pragma unroll
  for (int v = 0; v < 8; ++v) { cP[v] = bp; cQ[v] = 0.0f; }
  for (int k0 = 0; k0 < Hd; k0 += 4) {
    v2f a  = load_a(h, Hd, rowBase, k0, lane16, laneHalf);
    v2f bA = load_b(w1l, Hd, k0, colBase, lane16, laneHalf);
    v2f bB = load_b(w1l + Hd * Hd, Hd, k0, colBase, lane16, laneHalf);
    cP = wmma_f32(a, bA, cP);
    cQ = wmma_f32(a, bB, cQ);
  }
#pragma unroll
  for (int v = 0; v < 8; ++v) {
    int r = rowBase + v + 8 * laneHalf;
    int c = colBase + lane16;
    P[(size_t)r * Hd + c] = cP[v];
    Q[(size_t)r * Hd + c] = cQ[v];
  }
}

// ---------------------------------------------------------------------------
// S[b,i,:] = sum_{j!=i} silu(P[b,i,:] + Q[b,j,:] + d2(i,j) * w1_d[:])
__global__ __launch_bounds__(Hd) void silu_sum_kernel(
    const float* __restrict__ state, const float* __restrict__ P,
    const float* __restrict__ Q, const float* __restrict__ w1d,
    float* __restrict__ S) {
  int row = blockIdx.x;            // b*256 + i
  int b = row >> 8, i = row & 255;
  int t = threadIdx.x;             // channel
  __shared__ float px[Nn], py[Nn];
  for (int j = t; j < Nn; j += blockDim.x) {
    const float* s = state + (size_t)(b * Nn + j) * 4;
    px[j] = s[0]; py[j] = s[1];
  }
  __syncthreads();
  float pix = px[i], piy = py[i];
  float p  = P[(size_t)row * Hd + t];
  float wd = w1d[t];
  const float* Qb = Q + (size_t)b * Nn * Hd + t;
  float acc = 0.0f;
  for (int j = 0; j < Nn; ++j) {
    float dx = pix - px[j], dy = piy - py[j];
    float d2 = fmaf(dx, dx, dy * dy);
    float x  = fmaf(d2, wd, p + Qb[(size_t)j * Hd]);
    float sv = x / (1.0f + __expf(-x));      // silu -> v_exp_f32 + v_rcp_f32
    acc += (j == i) ? 0.0f : sv;
  }
  S[(size_t)row * Hd + t] = acc;
}

// ---------------------------------------------------------------------------
// AGG = S @ w2 + 255*b2
__global__ __launch_bounds__(128) void agg_gemm_kernel(
    const float* __restrict__ S, const float* __restrict__ w2l,
    const float* __restrict__ b2l, float* __restrict__ AGG) {
  int wave = (blockIdx.x * blockDim.x + threadIdx.x) >> 5;
  int lane = threadIdx.x & 31;
  int lane16 = lane & 15, laneHalf = lane >> 4;
  int rowBase = (wave >> 3) * 16;
  int colBase = (wave & 7) * 16;
  v8f c;
  float bias = 255.0f * b2l[colBase + lane16];
#pragma unroll
  for (int v = 0; v < 8; ++v) c[v] = bias;
  for (int k0 = 0; k0 < Hd; k0 += 4) {
    v2f a = load_a(S, Hd, rowBase, k0, lane16, laneHalf);
    v2f b = load_b(w2l, Hd, k0, colBase, lane16, laneHalf);
    c = wmma_f32(a, b, c);
  }
#pragma unroll
  for (int v = 0; v < 8; ++v)
    AGG[(size_t)(rowBase + v + 8 * laneHalf) * Hd + colBase + lane16] = c[v];
}

// ---------------------------------------------------------------------------
// h_out = h_in + h_in @ wu[0:128] + AGG @ wu[128:256] + bu
__global__ __launch_bounds__(128) void update_gemm_kernel(
    const float* __restrict__ hin, const float* __restrict__ AGG,
    const float* __restrict__ wul, const float* __restrict__ bul,
    float* __restrict__ hout) {
  int wave = (blockIdx.x * blockDim.x + threadIdx.x) >> 5;
  int lane = threadIdx.x & 31;
  int lane16 = lane & 15, laneHalf = lane >> 4;
  int rowBase = (wave >> 3) * 16;
  int colBase = (wave & 7) * 16;
  v8f c;
  float bias = bul[colBase + lane16];
#pragma unroll
  for (int v = 0; v < 8; ++v) c[v] = bias;
  for (int k0 = 0; k0 < Hd; k0 += 4) {   // h @ wu_top
    v2f a = load_a(hin, Hd, rowBase, k0, lane16, laneHalf);
    v2f b = load_b(wul, Hd, k0, colBase, lane16, laneHalf);
    c = wmma_f32(a, b, c);
  }
  for (int k0 = 0; k0 < Hd; k0 += 4) {   // agg @ wu_bot
    v2f a = load_a(AGG, Hd, rowBase, k0, lane16, laneHalf);
    v2f b = load_b(wul + Hd * Hd, Hd, k0, colBase, lane16, laneHalf);
    c = wmma_f32(a, b, c);
  }
#pragma unroll
  for (int v = 0; v < 8; ++v) {
    size_t idx = (size_t)(rowBase + v + 8 * laneHalf) * Hd + colBase + lane16;
    hout[idx] = hin[idx] + c[v];         // residual
  }
}

// ---------------------------------------------------------------------------
// out = state + h @ out_w + out_b   (N-dim = 4, trivial)
__global__ __launch_bounds__(128) void out_kernel(
    const float* __restrict__ state, const float* __restrict__ h,
    const float* __restrict__ ow, const float* __restrict__ ob,
    float* __restrict__ out) {
  int idx = blockIdx.x * blockDim.x + threadIdx.x;   // 0..8191
  if (idx >= ROWS * 4) return;
  int row = idx >> 2, c = idx & 3;
  const float* hr = h + (size_t)row * Hd;
  float acc = ob[c];
#pragma unroll 4
  for (int k = 0; k < Hd; ++k) acc = fmaf(hr[k], ow[k * 4 + c], acc);
  out[idx] = state[idx] + acc;
}

// ---------------------------------------------------------------------------
extern "C" void kernel_launch(void* const* d_in, const int* in_sizes, int n_in,
                              void* d_out, int out_size, void* d_ws,
                              size_t ws_size, hipStream_t stream) {
  const float* state   = (const float*)d_in[0];
  const float* embed_w = (const float*)d_in[1];
  const float* embed_b = (const float*)d_in[2];
  const float* w1      = (const float*)d_in[3];   // (4, 257, 128)
  const float* b1      = (const float*)d_in[4];
  const float* w2      = (const float*)d_in[5];   // (4, 128, 128)
  const float* b2      = (const float*)d_in[6];
  const float* wu      = (const float*)d_in[7];   // (4, 256, 128)
  const float* bu      = (const float*)d_in[8];
  const float* out_w   = (const float*)d_in[9];
  const float* out_b   = (const float*)d_in[10];
  float* out = (float*)d_out;

  // Workspace: 6 buffers of 2048x128 f32 = 6 MB total.
  const size_t RH = (size_t)ROWS * Hd;
  float* ws  = (float*)d_ws;
  float* h0  = ws;
  float* h1  = ws + RH;
  float* P   = ws + 2 * RH;
  float* Q   = ws + 3 * RH;
  float* S   = ws + 4 * RH;
  float* AGG = ws + 5 * RH;

  embed_kernel<<<ROWS, Hd, 0, stream>>>(state, embed_w, embed_b, h0);

  float* hin = h0;
  float* hout = h1;
  for (int l = 0; l < 4; ++l) {
    const float* w1l = w1 + (size_t)l * 257 * Hd;
    pq_gemm_kernel<<<256, 128, 0, stream>>>(hin, w1l, b1 + l * Hd, P, Q);
    silu_sum_kernel<<<ROWS, Hd, 0, stream>>>(state, P, Q, w1l + 256 * Hd, S);
    agg_gemm_kernel<<<256, 128, 0, stream>>>(S, w2 + (size_t)l * Hd * Hd,
                                             b2 + l * Hd, AGG);
    update_gemm_kernel<<<256, 128, 0, stream>>>(hin, AGG,
                                                wu + (size_t)l * 2 * Hd * Hd,
                                                bu + l * Hd, hout);
    float* tmp = hin; hin = hout; hout = tmp;
  }
  // L=4 swaps -> final h is back in h0 (== hin).
  out_kernel<<<(ROWS * 4) / 128, 128, 0, stream>>>(state, hin, out_w, out_b,
                                                   out);
}